// DDFlowLoss_15461882266080
// MI455X (gfx1250) — compile-verified
//
#include <hip/hip_runtime.h>
#include <cstdint>

// Problem constants (reference setup_inputs: B=8, H=384, W=768)
#define B_   8
#define H_   384
#define W_   768
#define HW_  (H_ * W_)          // 294912
#define HP_  (H_ + 6)           // 390  (3-px zero halo each side for 7x7 census)
#define WP_  (W_ + 6)           // 774
#define PP_  (HP_ * WP_)        // padded plane per batch
#define NB_  (B_ * PP_)         // one padded plane set across batch
#define BHW_ (B_ * HW_)

// ws layout (floats):
//  [0..15]                accumulators: 0=photo_fw 1=photo_bw 2=msum_fw 3=msum_bw 4=census_fw 5=census_bw
//  [16 .. 16+4*NB_)       padded gray planes: g1, gw2, g2, gw1   (zero borders)
//  [16+4*NB_ .. +2*BHW_)  masks: mask_fw, mask_bw
// total = 16 + 4*NB_ + 2*BHW_ = 14,378,128 floats ~= 57.5 MB

typedef float        v2f  __attribute__((ext_vector_type(2)));
typedef float        v8f  __attribute__((ext_vector_type(8)));
typedef unsigned int u32x4 __attribute__((ext_vector_type(4)));
typedef int          i32x8 __attribute__((ext_vector_type(8)));
typedef int          i32x4 __attribute__((ext_vector_type(4)));

// ---------------------------------------------------------------- utilities
__device__ __forceinline__ int imin_(int a, int b) { return a < b ? a : b; }

// (x)^0.4 for x > 0: raw v_log_f32 + v_exp_f32 (no NaN/denorm special-casing)
__device__ __forceinline__ float pow04_(float x) {
  return __builtin_amdgcn_exp2f(0.4f * __builtin_amdgcn_logf(x));
}

// lane-xor exchange via ds_swizzle (group-of-32: and=0x1f, or=0, xor=MASK)
template <int MASK>
__device__ __forceinline__ float xor_add_(float v) {
  int s = __builtin_amdgcn_ds_swizzle(__float_as_int(v), (MASK << 10) | 0x1f);
  return v + __int_as_float(s);
}

template <int C>
__device__ __forceinline__ void bilinear_warp(const float* __restrict__ p, int x, int y,
                                              float fx, float fy, float* out) {
  float gx = fminf(fmaxf((float)x + fx, 0.0f), (float)(W_ - 1));
  float gy = fminf(fmaxf((float)y + fy, 0.0f), (float)(H_ - 1));
  float x0f = floorf(gx), y0f = floorf(gy);
  float wx = gx - x0f,    wy = gy - y0f;
  int x0 = (int)x0f, y0 = (int)y0f;
  int x1 = imin_(x0 + 1, W_ - 1), y1 = imin_(y0 + 1, H_ - 1);
  int i00 = y0 * W_ + x0, i01 = y0 * W_ + x1;
  int i10 = y1 * W_ + x0, i11 = y1 * W_ + x1;
  float w00 = (1.f - wx) * (1.f - wy), w01 = wx * (1.f - wy);
  float w10 = (1.f - wx) * wy,         w11 = wx * wy;
#pragma unroll
  for (int c = 0; c < C; ++c) {
    const float* pc = p + (size_t)c * HW_;
    out[c] = pc[i00] * w00 + pc[i01] * w01 + pc[i10] * w10 + pc[i11] * w11;
  }
}

// ---------------------------------------------------------------- init
__global__ void zero_ws(float* ws, int n) {
  int i = blockIdx.x * blockDim.x + threadIdx.x;
  int stride = gridDim.x * blockDim.x;
  for (; i < n; i += stride) ws[i] = 0.0f;
}

// ---------------------------------------------------------------- pass 1
// Fused: warps, occlusion, photometric robust loss partials, gray planes, masks.
__global__ void pass1(const float* __restrict__ img1, const float* __restrict__ img2,
                      const float* __restrict__ ffw,  const float* __restrict__ fbw,
                      float* __restrict__ occ_out, float* __restrict__ ws) {
  int i = blockIdx.x * 256 + threadIdx.x;          // grid exactly covers B*H*W
  int b = i / HW_;
  int r = i - b * HW_;
  int y = r / W_;
  int x = r - y * W_;

  const float* I1 = img1 + (size_t)b * 3 * HW_;
  const float* I2 = img2 + (size_t)b * 3 * HW_;
  const float* F1 = ffw  + (size_t)b * 2 * HW_;
  const float* F2 = fbw  + (size_t)b * 2 * HW_;

  float f1x = F1[r], f1y = F1[HW_ + r];            // flow_fw at pixel
  float f2x = F2[r], f2y = F2[HW_ + r];            // flow_bw at pixel

  float w2[3], w1[3], fbww[2], ffww[2];
  bilinear_warp<3>(I2, x, y, f1x, f1y, w2);        // img2 warped by flow_fw
  bilinear_warp<3>(I1, x, y, f2x, f2y, w1);        // img1 warped by flow_bw
  bilinear_warp<2>(F2, x, y, f1x, f1y, fbww);      // flow_bw warped by flow_fw
  bilinear_warp<2>(F1, x, y, f2x, f2y, ffww);      // flow_fw warped by flow_bw

  // occlusion masks
  float dfx = f1x + fbww[0], dfy = f1y + fbww[1];
  float mag_f = dfx * dfx + dfy * dfy;
  float fm_f  = f1x * f1x + f1y * f1y + fbww[0] * fbww[0] + fbww[1] * fbww[1];
  float occf  = (mag_f > 0.01f * fm_f + 0.5f) ? 1.0f : 0.0f;

  float dbx = f2x + ffww[0], dby = f2y + ffww[1];
  float mag_b = dbx * dbx + dby * dby;
  float fm_b  = f2x * f2x + f2y * f2y + ffww[0] * ffww[0] + ffww[1] * ffww[1];
  float occb  = (mag_b > 0.01f * fm_b + 0.5f) ? 1.0f : 0.0f;

  occ_out[i] = occf;                               // d_out[3 + i]
  float mf = 1.0f - occf, mb = 1.0f - occb;

  float i1r = I1[r], i1g = I1[HW_ + r], i1b = I1[2 * HW_ + r];
  float i2r = I2[r], i2g = I2[HW_ + r], i2b = I2[2 * HW_ + r];

  // photometric robust loss partials: sum_c (|diff|+0.01)^0.4 * mask
  float pnf = (pow04_(fabsf(i1r - w2[0]) + 0.01f) +
               pow04_(fabsf(i1g - w2[1]) + 0.01f) +
               pow04_(fabsf(i1b - w2[2]) + 0.01f)) * mf;
  float pnb = (pow04_(fabsf(i2r - w1[0]) + 0.01f) +
               pow04_(fabsf(i2g - w1[1]) + 0.01f) +
               pow04_(fabsf(i2b - w1[2]) + 0.01f)) * mb;

  // gray intensities * 255 into zero-padded planes (census input)
  float g1  = (0.2989f * i1r   + 0.587f * i1g   + 0.114f * i1b)   * 255.0f;
  float g2  = (0.2989f * i2r   + 0.587f * i2g   + 0.114f * i2b)   * 255.0f;
  float gw2 = (0.2989f * w2[0] + 0.587f * w2[1] + 0.114f * w2[2]) * 255.0f;
  float gw1 = (0.2989f * w1[0] + 0.587f * w1[1] + 0.114f * w1[2]) * 255.0f;

  float* planes = ws + 16;
  size_t pp = (size_t)b * PP_ + (size_t)(y + 3) * WP_ + (x + 3);
  planes[pp]                   = g1;
  planes[NB_ + pp]             = gw2;
  planes[2 * (size_t)NB_ + pp] = g2;
  planes[3 * (size_t)NB_ + pp] = gw1;

  float* masks = ws + 16 + 4 * (size_t)NB_;
  masks[i]        = mf;
  masks[BHW_ + i] = mb;

  // wave32 reduction of 4 scalars via ds_swizzle xor patterns
  float smf = mf, smb = mb;
  pnf = xor_add_<1>(pnf);  pnf = xor_add_<2>(pnf);  pnf = xor_add_<4>(pnf);
  pnf = xor_add_<8>(pnf);  pnf = xor_add_<16>(pnf);
  pnb = xor_add_<1>(pnb);  pnb = xor_add_<2>(pnb);  pnb = xor_add_<4>(pnb);
  pnb = xor_add_<8>(pnb);  pnb = xor_add_<16>(pnb);
  smf = xor_add_<1>(smf);  smf = xor_add_<2>(smf);  smf = xor_add_<4>(smf);
  smf = xor_add_<8>(smf);  smf = xor_add_<16>(smf);
  smb = xor_add_<1>(smb);  smb = xor_add_<2>(smb);  smb = xor_add_<4>(smb);
  smb = xor_add_<8>(smb);  smb = xor_add_<16>(smb);

  __shared__ float blk[4];
  if (threadIdx.x < 4) blk[threadIdx.x] = 0.0f;
  __syncthreads();
  if ((threadIdx.x & 31) == 0) {
    atomicAdd(&blk[0], pnf);
    atomicAdd(&blk[1], pnb);
    atomicAdd(&blk[2], smf);
    atomicAdd(&blk[3], smb);
  }
  __syncthreads();
  if (threadIdx.x < 4) atomicAdd(&ws[threadIdx.x], blk[threadIdx.x]);
}

// ---------------------------------------------------------------- TDM tile load
// 2D tile (38 x 14 floats) from a padded gray plane into LDS, via the Tensor
// Data Mover.  D# per CDNA5 ISA ch.8 (group0: count/lds_addr/global_addr/type,
// group1: data_size=4B, tensor 774x390 stride 774, tile 38x14).
// This toolchain exposes the 6-arg builtin:
//   (uint32x4 g0, int32x8 g1, int32x4 g2, int32x4 g3, int32x8 extra, i32 cpol)
__device__ __forceinline__ void tdm_load_tile(uint64_t gaddr, uint32_t lds_byte_off) {
  u32x4 g0;
  g0[0] = 1u;                                                  // count=1, no gather
  g0[1] = lds_byte_off;                                        // lds_addr
  g0[2] = (uint32_t)(gaddr & 0xffffffffull);                   // global_addr[31:0]
  g0[3] = ((uint32_t)(gaddr >> 32) & 0x01ffffffu) | 0x80000000u; // addr[56:32] | type=2
  i32x8 g1;
  g1[0] = (int)(2u << 16);            // data_size = 4 bytes; no flags
  g1[1] = (int)((uint32_t)WP_ << 16); // tensor_dim0 = 774 (low 16)
  g1[2] = (int)((uint32_t)HP_ << 16); // tensor_dim1 = 390 (low 16)
  g1[3] = (int)(38u << 16);           // tile_dim0 = 38
  g1[4] = 14;                         // tile_dim1 = 14, tile_dim2 = 0
  g1[5] = WP_;                        // tensor_dim0_stride = 774
  g1[6] = 0;                          // stride hi / dim1_stride lo
  g1[7] = 0;
  i32x4 g2; g2[0] = 1; g2[1] = 1; g2[2] = 0; g2[3] = 0;  // tensor_dim2=1, tensor_dim3=1
  i32x4 g3; g3[0] = 0; g3[1] = (1 << 16); g3[2] = 0; g3[3] = 0; // tensor_dim4=1
  i32x8 gz; gz[0]=0; gz[1]=0; gz[2]=0; gz[3]=0; gz[4]=0; gz[5]=0; gz[6]=0; gz[7]=0;
  __builtin_amdgcn_tensor_load_to_lds(g0, g1, g2, g3, gz, 0);
}

// ---------------------------------------------------------------- pass 2: census
// grid: (W/32, (H/8)*B, 2)  block: (32,8).  z=0: fw (g1 vs gw2, mask_fw),
// z=1: bw (g2 vs gw1, mask_bw).
__global__ void pass2(float* __restrict__ ws) {
  __shared__ float T1[14 * 38];
  __shared__ float T2[14 * 38];

  const int z  = blockIdx.z;
  const int bx = blockIdx.x;
  const int by = blockIdx.y % (H_ / 8);
  const int b  = blockIdx.y / (H_ / 8);
  const int x0 = bx * 32, y0 = by * 8;

  const float* planes = ws + 16;
  const float* P1 = planes + (size_t)(z ? 2 : 0) * NB_ + (size_t)b * PP_;
  const float* P2 = planes + (size_t)(z ? 3 : 1) * NB_ + (size_t)b * PP_;

  if (threadIdx.y == 0) {              // wave 0 only (wave32, uniform branch)
    uint64_t ga1 = (uint64_t)(uintptr_t)(P1 + (size_t)y0 * WP_ + x0);
    uint64_t ga2 = (uint64_t)(uintptr_t)(P2 + (size_t)y0 * WP_ + x0);
    tdm_load_tile(ga1, (uint32_t)(uintptr_t)(void*)T1);
    tdm_load_tile(ga2, (uint32_t)(uintptr_t)(void*)T2);
    __builtin_amdgcn_s_wait_tensorcnt(0);
  }
  __syncthreads();

  const int lx = threadIdx.x, ly = threadIdx.y;
  float c1 = T1[(ly + 3) * 38 + lx + 3];
  float c2 = T2[(ly + 3) * 38 + lx + 3];

  float s = 0.0f;
#pragma unroll
  for (int dy = 0; dy < 7; ++dy) {
#pragma unroll
    for (int dx = 0; dx < 7; ++dx) {
      float t1 = T1[(ly + dy) * 38 + lx + dx] - c1;
      float t2 = T2[(ly + dy) * 38 + lx + dx] - c2;
      float f1 = t1 * __builtin_amdgcn_rsqf(0.81f + t1 * t1);  // v_rsq_f32
      float f2 = t2 * __builtin_amdgcn_rsqf(0.81f + t2 * t2);
      float d  = (f1 - f2) * (f1 - f2);
      s += d * __builtin_amdgcn_rcpf(0.1f + d);                // v_rcp_f32
    }
  }

  const float* masks = planes + 4 * (size_t)NB_;
  float m = masks[(size_t)z * BHW_ + (size_t)b * HW_ + (size_t)(y0 + ly) * W_ + (x0 + lx)];
  float rr = pow04_(s + 0.01f) * m;

  // Wave-sum via V_WMMA_F32_16X16X4_F32: A = ones(16x4), B holds one partial
  // per lane (B[k,n] slots), so D[m,n] = sum_k B[k,n] = p[n] + p[n+16].
  // EXEC is all-ones here (full 32x8 block, converged).
  v2f a;  a.x = 1.0f; a.y = 1.0f;
  v2f bm; bm.x = rr;  bm.y = 0.0f;
  v8f cz = {};
  cz = __builtin_amdgcn_wmma_f32_16x16x4_f32(false, a, false, bm, (short)0, cz,
                                             false, false);
  float t = cz[0];                      // lane L: p[L&15] + p[(L&15)+16]
  t = xor_add_<1>(t);
  t = xor_add_<2>(t);
  t = xor_add_<4>(t);
  t = xor_add_<8>(t);                   // full wave sum, replicated

  __shared__ float blk;
  if (threadIdx.x == 0 && threadIdx.y == 0) blk = 0.0f;
  __syncthreads();
  if (threadIdx.x == 0) atomicAdd(&blk, t);
  __syncthreads();
  if (threadIdx.x == 0 && threadIdx.y == 0) atomicAdd(&ws[4 + z], blk);
}

// ---------------------------------------------------------------- finalize
__global__ void finalize(const float* __restrict__ ws, float* __restrict__ out) {
  float pnf = ws[0], pnb = ws[1], msf = ws[2], msb = ws[3], cnf = ws[4], cnb = ws[5];
  float df = 2.0f * msf + 1e-6f, db = 2.0f * msb + 1e-6f;
  float photo  = pnf / df + pnb / db;
  float census = cnf / df + cnb / db;
  out[0] = photo + census;
  out[1] = photo;
  out[2] = census;
}

// ---------------------------------------------------------------- launch
extern "C" void kernel_launch(void* const* d_in, const int* in_sizes, int n_in,
                              void* d_out, int out_size, void* d_ws, size_t ws_size,
                              hipStream_t stream) {
  const float* img1 = (const float*)d_in[0];
  const float* img2 = (const float*)d_in[1];
  const float* ffw  = (const float*)d_in[2];
  const float* fbw  = (const float*)d_in[3];
  float* out = (float*)d_out;
  float* ws  = (float*)d_ws;

  // zero accumulators + padded gray planes (borders must be 0 for census pad)
  int nz = 16 + 4 * NB_;
  zero_ws<<<2048, 256, 0, stream>>>(ws, nz);

  pass1<<<BHW_ / 256, 256, 0, stream>>>(img1, img2, ffw, fbw, out + 3, ws);

  dim3 g2(W_ / 32, (H_ / 8) * B_, 2), b2(32, 8);
  pass2<<<g2, b2, 0, stream>>>(ws);

  finalize<<<1, 1, 0, stream>>>(ws, out);
}